// SlidingKVCacheManager_82978768159599
// MI455X (gfx1250) — compile-verified
//
#include <hip/hip_runtime.h>
#include <stdint.h>

// SKVQ fake-quant of KV cache: per-token-row (head_dim=128) 2-bit asymmetric
// min/max fake quantization for the "mid" token range; sink/window copied.
// Pure streaming kernel: ~1.07 GB of traffic, HBM-bound (~46 us at 23.3 TB/s).
// wave32: one wave per 128-float row, 4 floats per lane, shfl_xor reductions.
// Data movement uses the gfx1250 async global->LDS DMA path (ASYNCcnt).

typedef float v4f __attribute__((ext_vector_type(4)));
typedef int   v4i __attribute__((ext_vector_type(4)));

#define AS1 __attribute__((address_space(1)))
#define AS3 __attribute__((address_space(3)))

typedef AS1 v4i* gptr_b128;   // global b128 source (as printed by clang: v4i __device__*)
typedef AS3 v4i* lptr_b128;   // LDS b128 destination

static constexpr int   HD       = 128;   // head_dim
static constexpr int   TOK      = 8192;  // tokens (power of two -> cheap mask)
static constexpr int   SINK     = 32;
static constexpr int   WINDOW   = 128;
static constexpr float QMAX     = 3.0f;  // 2^2 - 1
static constexpr int   WAVES_PB = 8;     // 256 threads / 32
static constexpr int   DEPTH    = 2;     // async double buffer

#if __has_builtin(__builtin_amdgcn_global_load_async_to_lds_b128) && \
    __has_builtin(__builtin_amdgcn_s_wait_asynccnt)
#define HAVE_ASYNC_LDS 1
#else
#define HAVE_ASYNC_LDS 0
#endif

__device__ __forceinline__ float wave_min32(float v) {
#pragma unroll
  for (int m = 16; m >= 1; m >>= 1) v = fminf(v, __shfl_xor(v, m, 32));
  return v;
}
__device__ __forceinline__ float wave_max32(float v) {
#pragma unroll
  for (int m = 16; m >= 1; m >>= 1) v = fmaxf(v, __shfl_xor(v, m, 32));
  return v;
}

// x: this lane's 4 contiguous elements of the row. t: token index (wave-uniform).
__device__ __forceinline__ void process_row(v4f x, int t,
                                            float* __restrict__ dst, int lane) {
  const bool quant = (t >= SINK) & (t < TOK - WINDOW);
  if (quant) {
    float lmn = fminf(fminf(x.x, x.y), fminf(x.z, x.w));
    float lmx = fmaxf(fmaxf(x.x, x.y), fmaxf(x.z, x.w));
    float mn = wave_min32(lmn);
    float mx = wave_max32(lmx);
    float scale = fmaxf((mx - mn) * (1.0f / QMAX), 1e-8f);
    float inv   = 1.0f / scale;
    float zp    = rintf(-mn * inv);          // round-to-nearest-even
#pragma unroll
    for (int i = 0; i < 4; ++i) {
      float q = fminf(fmaxf(rintf(x[i] * inv) + zp, 0.0f), QMAX);
      x[i] = (q - zp) * scale;
    }
  }
  __builtin_nontemporal_store(x, (v4f*)dst + lane);  // th:NT streaming store
}

__global__ void __launch_bounds__(256) skvq_kernel(
    const float* __restrict__ kf, const float* __restrict__ vf,
    float* __restrict__ out, long long rows_per_tensor) {
  const int lane = threadIdx.x & 31;
  const int wave = threadIdx.x >> 5;
  const long long total_rows = 2 * rows_per_tensor;
  const long long wstride = (long long)gridDim.x * WAVES_PB;
  const long long wid = (long long)blockIdx.x * WAVES_PB + wave;

#if HAVE_ASYNC_LDS
  // Double-buffered async global->LDS pipeline (ASYNCcnt-tracked DMA path).
  __shared__ __align__(16) float lds[WAVES_PB * DEPTH * HD];
  float* buf = &lds[wave * (DEPTH * HD)];

  auto issue = [&](long long row, int slot) {
    const long long rin = (row < rows_per_tensor) ? row : row - rows_per_tensor;
    const float* src = (row < rows_per_tensor) ? kf : vf;
    __builtin_amdgcn_global_load_async_to_lds_b128(
        (gptr_b128)(src + rin * HD + lane * 4),
        (lptr_b128)(buf + slot * HD + lane * 4),
        /*offset=*/0, /*cpol=*/0);
  };

  if (wid < total_rows) issue(wid, 0);
  int slot = 0;
  for (long long r = wid; r < total_rows; r += wstride) {
    const long long rn = r + wstride;
    if (rn < total_rows) {
      issue(rn, slot ^ 1);                       // prefetch next row
      __builtin_amdgcn_s_wait_asynccnt(1);       // wait for current row only
    } else {
      __builtin_amdgcn_s_wait_asynccnt(0);
    }
    asm volatile("" ::: "memory");               // keep LDS reads below the wait
    v4f x = *((const v4f*)(buf + slot * HD) + lane);
    const long long rin = (r < rows_per_tensor) ? r : r - rows_per_tensor;
    process_row(x, (int)(rin & (TOK - 1)), out + r * HD, lane);
    slot ^= 1;
  }
#else
  for (long long r = wid; r < total_rows; r += wstride) {
    const long long rin = (r < rows_per_tensor) ? r : r - rows_per_tensor;
    const float* src = (r < rows_per_tensor) ? kf : vf;
    const v4f* in4 = (const v4f*)(src + rin * HD);
    // Speculative L2 prefetch of this wave's next row (global_prefetch_b8).
    __builtin_prefetch((const char*)(in4 + lane) + (size_t)wstride * HD * 4, 0, 0);
    v4f x = __builtin_nontemporal_load(in4 + lane);  // th:NT streaming load
    process_row(x, (int)(rin & (TOK - 1)), out + r * HD, lane);
  }
#endif
}

extern "C" void kernel_launch(void* const* d_in, const int* in_sizes, int n_in,
                              void* d_out, int out_size, void* d_ws, size_t ws_size,
                              hipStream_t stream) {
  (void)n_in; (void)out_size; (void)d_ws; (void)ws_size;
  const float* kf = (const float*)d_in[0];
  const float* vf = (const float*)d_in[1];
  float* out = (float*)d_out;

  const long long n = (long long)in_sizes[0];      // 2*32*8192*128
  const long long rows_per_tensor = n / HD;        // 524288
  const long long total_rows = 2 * rows_per_tensor;

  long long blocks = (total_rows + WAVES_PB - 1) / WAVES_PB;
  if (blocks > 16384) blocks = 16384;              // grid-stride, ~8 rows/wave
  skvq_kernel<<<dim3((unsigned)blocks), 32 * WAVES_PB, 0, stream>>>(
      kf, vf, out, rows_per_tensor);
}